// Dilate_84902913507553
// MI455X (gfx1250) — compile-verified
//
#include <hip/hip_runtime.h>
#include <hip/hip_bf16.h>
#include <stdint.h>

// Morphological 3x3 dilation (all-ones kernel, border = -2.0) over
// (16, 8, 512, 512) f32. Pure HBM-bandwidth problem (~268 MB total traffic,
// ~11.5 us at 23.3 TB/s). Strategy: gfx1250 async DMA (global -> LDS) stages
// each input byte ~once (1.125x halo amplification, L2-absorbed), separable
// max computed from LDS, non-temporal coalesced 128-bit stores. WMMA is not
// applicable (max is not a MAC); the CDNA5 paths exercised are the async
// global->LDS pipeline + split wait counters.

#define IMG_H   512
#define IMG_W   512
#define TILE_H  16            // output rows per block
#define NPLANES (16 * 8)      // N * C
#define BORDER  (-2.0f)

typedef float v4f __attribute__((ext_vector_type(4)));

// ---- gfx1250 async global->LDS DMA (ASYNCcnt-tracked) ----------------------
// GVS addressing: mem_addr = SGPR64 base + VGPR32 offset. LDS destination is
// the per-lane byte offset into the workgroup's LDS allocation.
__device__ __forceinline__ void async_ld_b128(uint32_t lds_byte,
                                              uint32_t glob_byte_off,
                                              const float* base) {
  asm volatile("global_load_async_to_lds_b128 %0, %1, %2"
               :
               : "v"(lds_byte), "v"(glob_byte_off), "s"(base)
               : "memory");
}

__device__ __forceinline__ void wait_async_zero() {
#if defined(__has_builtin)
#if __has_builtin(__builtin_amdgcn_s_wait_asynccnt)
  __builtin_amdgcn_s_wait_asynccnt(0);
  return;
#endif
#endif
  asm volatile("s_wait_asynccnt 0x0" ::: "memory");
}

// Horizontal 3-max for a 4-wide quad starting at x0 of one staged row.
__device__ __forceinline__ v4f hmax_row(const float* __restrict__ row,
                                        int x0) {
  v4f e = *reinterpret_cast<const v4f*>(row + x0);
  float l = (x0 == 0) ? BORDER : row[x0 - 1];
  float r = (x0 + 4 >= IMG_W) ? BORDER : row[x0 + 4];
  v4f h;
  h.x = fmaxf(fmaxf(l,   e.x), e.y);
  h.y = fmaxf(fmaxf(e.x, e.y), e.z);
  h.z = fmaxf(fmaxf(e.y, e.z), e.w);
  h.w = fmaxf(fmaxf(e.z, e.w), r);
  return h;
}

__global__ __launch_bounds__(256) void
Dilate_84902913507553_kernel(const float* __restrict__ img,
                             float* __restrict__ out) {
  // 18 rows x 512 f32 = 36 KB of LDS (input strip + 1-row halo each side).
  __shared__ float tile[(TILE_H + 2) * IMG_W];

  const int tid   = threadIdx.x;
  const int plane = blockIdx.x >> 5;      // IMG_H / TILE_H == 32 strips/plane
  const int strip = blockIdx.x & 31;
  const int y0    = strip * TILE_H;
  const uint32_t plane_off = (uint32_t)plane * (IMG_H * IMG_W); // elements

  // ---- Stage (TILE_H+2) x 512 rows: gy = y0-1 .. y0+TILE_H -----------------
  // 18 rows * 128 float4 = 2304 DMA transfers; 9 per thread. The gy bound
  // check is wave-uniform (r is constant across each wave's 32 lanes).
#pragma unroll
  for (int i = 0; i < ((TILE_H + 2) * (IMG_W / 4)) / 256; ++i) {
    const int idx   = i * 256 + tid;      // float4 index within the tile
    const int r     = idx >> 7;           // 128 float4 per row
    const int xq    = idx & 127;
    const int gy    = y0 - 1 + r;
    const int lds_f = r * IMG_W + xq * 4; // float index in LDS
    if ((unsigned)gy < (unsigned)IMG_H) {
      const uint32_t goff =
          (plane_off + (uint32_t)gy * IMG_W + (uint32_t)(xq * 4)) * 4u;
      async_ld_b128((uint32_t)(lds_f * 4), goff, img);
    } else {
      // Halo row outside the image: border value (faithful to -2.0 padding).
      v4f bv = {BORDER, BORDER, BORDER, BORDER};
      *reinterpret_cast<v4f*>(&tile[lds_f]) = bv;
    }
  }
  wait_async_zero();   // this wave's DMA done (LDS writes landed)
  __syncthreads();     // all waves' staging visible

  // ---- Compute: separable 3x3 max ------------------------------------------
  // thread -> (column quad, row half). 128 quads cover the 512-wide row;
  // each half walks 8 output rows with a rolling vertical 3-max.
  const int quad  = tid & 127;
  const int half  = tid >> 7;
  const int x0    = quad * 4;
  const int orow0 = half * (TILE_H / 2);  // 0 or 8

  // LDS row r corresponds to gy = y0 - 1 + r; output row (orow0 + o) needs
  // LDS rows (orow0 + o) .. (orow0 + o + 2).
  v4f a = hmax_row(&tile[(orow0 + 0) * IMG_W], x0);
  v4f b = hmax_row(&tile[(orow0 + 1) * IMG_W], x0);

  uint32_t obase = plane_off + (uint32_t)(y0 + orow0) * IMG_W + (uint32_t)x0;
#pragma unroll
  for (int o = 0; o < TILE_H / 2; ++o) {
    v4f c = hmax_row(&tile[(orow0 + o + 2) * IMG_W], x0);
    v4f v;
    v.x = fmaxf(fmaxf(a.x, b.x), c.x);
    v.y = fmaxf(fmaxf(a.y, b.y), c.y);
    v.z = fmaxf(fmaxf(a.z, b.z), c.z);
    v.w = fmaxf(fmaxf(a.w, b.w), c.w);
    // Output is written once, never re-read: stream it past L2 (TH_STORE_NT)
    // so the cache keeps the cross-block halo rows instead.
    __builtin_nontemporal_store(
        v, reinterpret_cast<v4f*>(&out[obase + (uint32_t)o * IMG_W]));
    a = b;
    b = c;
  }
}

extern "C" void kernel_launch(void* const* d_in, const int* in_sizes, int n_in,
                              void* d_out, int out_size, void* d_ws,
                              size_t ws_size, hipStream_t stream) {
  (void)in_sizes; (void)n_in; (void)d_ws; (void)ws_size; (void)out_size;
  const float* img = (const float*)d_in[0];
  float* out = (float*)d_out;
  const int blocks = NPLANES * (IMG_H / TILE_H);  // 128 * 32 = 4096
  Dilate_84902913507553_kernel<<<blocks, 256, 0, stream>>>(img, out);
}